// GAT_DE_16045997818080
// MI455X (gfx1250) — compile-verified
//
#include <hip/hip_runtime.h>
#include <hip/hip_bf16.h>

typedef __attribute__((ext_vector_type(2))) float v2f;
typedef __attribute__((ext_vector_type(8))) float v8f;

#define IN_F   128
#define HID    64
#define NH     8
#define NEG_SLOPE 0.2f

// ---- monotone float <-> uint encoding for atomic max on floats ----
__device__ __forceinline__ unsigned enc_f(float f) {
    unsigned u = __float_as_uint(f);
    return (u & 0x80000000u) ? ~u : (u | 0x80000000u);
}
__device__ __forceinline__ float dec_f(unsigned e) {
    return (e & 0x80000000u) ? __uint_as_float(e & 0x7FFFFFFFu)
                             : __uint_as_float(~e);
}
// enc(-inf) = ~0xFF800000 = 0x007FFFFF
#define ENC_NEG_INF 0x007FFFFFu

// ---- 0: init output accumulator, per-(node,head) max and denom ----
__global__ __launch_bounds__(256) void init_kernel(float* __restrict__ out,
                                                   unsigned* __restrict__ m_enc,
                                                   float* __restrict__ denom, int n) {
    int i = blockIdx.x * blockDim.x + threadIdx.x;
    if (i < n * HID) out[i] = 0.0f;
    if (i < n * NH) { m_enc[i] = ENC_NEG_INF; denom[i] = 0.0f; }
}

// ---- 1: g = vert @ W via V_WMMA_F32_16X16X4_F32 ----
// 8 waves / block: wave w -> Mtile = 2*blockIdx + (w>>2), Ntile = w&3 (HID=64 -> 4 Ntiles)
__global__ __launch_bounds__(256) void gat_gemm_wmma(const float* __restrict__ vert,
                                                     const float* __restrict__ W,
                                                     float* __restrict__ g, int n) {
    const int wave = threadIdx.x >> 5;
    const int lane = threadIdx.x & 31;
    const int mtile = blockIdx.x * 2 + (wave >> 2);
    const int ntile = wave & 3;
    const int rlo   = lane & 15;
    const int khalf = lane >> 4;          // 0: K={0,1}, 1: K={2,3}
    const int col   = ntile * 16 + rlo;   // B/C column for this lane

    int row = mtile * 16 + rlo;
    int rowc = row < n ? row : n - 1;     // clamp loads; EXEC stays all-ones for WMMA

    const float* ap = vert + (long)rowc * IN_F + khalf * 2;
    const float* bp = W + (long)(khalf * 2) * HID + col;

    v8f acc = {};
#pragma unroll
    for (int kk = 0; kk < IN_F; kk += 4) {
        v2f a, b;
        a.x = ap[kk];
        a.y = ap[kk + 1];
        b.x = bp[(long)kk * HID];
        b.y = bp[(long)(kk + 1) * HID];
        acc = __builtin_amdgcn_wmma_f32_16x16x4_f32(
            /*neg_a=*/false, a, /*neg_b=*/false, b,
            /*c_mod=*/(short)0, acc, /*reuse_a=*/false, /*reuse_b=*/false);
    }

    // C layout: VGPR r -> M = r + 8*khalf, N = lane&15
    const int crow0 = mtile * 16 + khalf * 8;
#pragma unroll
    for (int r = 0; r < 8; ++r) {
        int rr = crow0 + r;
        if (rr < n) g[(long)rr * HID + col] = acc[r];
    }
}

// ---- 2: e_s[n,h] = g[n,h,:].a_src[h,:]; e_d likewise ----
__global__ __launch_bounds__(256) void edot_kernel(const float* __restrict__ g,
                                                   const float* __restrict__ a_src,
                                                   const float* __restrict__ a_dst,
                                                   float* __restrict__ e_s,
                                                   float* __restrict__ e_d, int n) {
    int i = blockIdx.x * blockDim.x + threadIdx.x;
    if (i >= n * NH) return;
    int h = i & 7;
    const float* gp = g + (long)(i >> 3) * HID + h * 8;
    float es = 0.f, ed = 0.f;
#pragma unroll
    for (int d = 0; d < 8; ++d) {
        float v = gp[d];
        es += v * a_src[h * 8 + d];
        ed += v * a_dst[h * 8 + d];
    }
    e_s[i] = es;
    e_d[i] = ed;
}

__device__ __forceinline__ float edge_score(const float* e_s, const float* e_d,
                                            int s, int d, int h) {
    float x = e_s[s * NH + h] + e_d[d * NH + h];
    return x > 0.f ? x : NEG_SLOPE * x;
}

// ---- 3: segment max over dst (ordered-uint atomic max) ----
__global__ __launch_bounds__(256) void edge_max_kernel(const int* __restrict__ src,
                                                       const int* __restrict__ dst,
                                                       const float* __restrict__ e_s,
                                                       const float* __restrict__ e_d,
                                                       unsigned* __restrict__ m_enc, int ne) {
    int i = blockIdx.x * blockDim.x + threadIdx.x;
    if (i >= ne * NH) return;
    int e = i >> 3, h = i & 7;
    int s = src[e], d = dst[e];
    float sc = edge_score(e_s, e_d, s, d, h);
    atomicMax(&m_enc[d * NH + h], enc_f(sc));
}

// ---- 4: denom = segment_sum(exp(score - m)) ----
__global__ __launch_bounds__(256) void edge_den_kernel(const int* __restrict__ src,
                                                       const int* __restrict__ dst,
                                                       const float* __restrict__ e_s,
                                                       const float* __restrict__ e_d,
                                                       const unsigned* __restrict__ m_enc,
                                                       float* __restrict__ denom, int ne) {
    int i = blockIdx.x * blockDim.x + threadIdx.x;
    if (i >= ne * NH) return;
    int e = i >> 3, h = i & 7;
    int s = src[e], d = dst[e];
    float sc = edge_score(e_s, e_d, s, d, h);
    float m = dec_f(m_enc[d * NH + h]);   // this dst has >=1 edge, so m > -inf
    unsafeAtomicAdd(&denom[d * NH + h], __expf(sc - m));
}

// ---- 5: out[dst] += alpha * g[src]; 16 threads/edge, 4 floats/thread ----
__global__ __launch_bounds__(256) void edge_agg_kernel(const int* __restrict__ src,
                                                       const int* __restrict__ dst,
                                                       const float* __restrict__ e_s,
                                                       const float* __restrict__ e_d,
                                                       const unsigned* __restrict__ m_enc,
                                                       const float* __restrict__ denom,
                                                       const float* __restrict__ g,
                                                       float* __restrict__ out, int ne) {
    int i = blockIdx.x * blockDim.x + threadIdx.x;
    if (i >= ne * 16) return;
    int e = i >> 4;
    int l = i & 15;       // 16 lanes cover 64 hidden dims, 4 each
    int h = l >> 1;       // 4 consecutive dims share one head (8 dims/head)
    int s = src[e], d = dst[e];
    float sc = edge_score(e_s, e_d, s, d, h);
    float m  = dec_f(m_enc[d * NH + h]);
    float ex = __expf(sc - m);
    float alpha = ex / fmaxf(denom[d * NH + h], 1e-16f);
    const float4 gv = *(const float4*)(g + (long)s * HID + l * 4);
    float* op = out + (long)d * HID + l * 4;
    unsafeAtomicAdd(op + 0, alpha * gv.x);
    unsafeAtomicAdd(op + 1, alpha * gv.y);
    unsafeAtomicAdd(op + 2, alpha * gv.z);
    unsafeAtomicAdd(op + 3, alpha * gv.w);
}

// ---- 6: in-place ELU ----
__global__ __launch_bounds__(256) void elu_kernel(float* __restrict__ out, int tot) {
    int i = blockIdx.x * blockDim.x + threadIdx.x;
    if (i >= tot) return;
    float x = out[i];
    out[i] = x > 0.f ? x : expm1f(x);
}

extern "C" void kernel_launch(void* const* d_in, const int* in_sizes, int n_in,
                              void* d_out, int out_size, void* d_ws, size_t ws_size,
                              hipStream_t stream) {
    const float* vert  = (const float*)d_in[0];
    const int*   edge  = (const int*)d_in[1];   // [2, E] int32 (JAX x64 disabled)
    const float* W     = (const float*)d_in[2]; // [128, 8, 8] == [128, 64] row-major
    const float* a_src = (const float*)d_in[3];
    const float* a_dst = (const float*)d_in[4];

    const int n  = in_sizes[0] / IN_F;
    const int ne = in_sizes[1] / 2;
    const int* src = edge;
    const int* dst = edge + ne;

    // workspace: g[N*64] | e_s[N*8] | e_d[N*8] | m_enc[N*8] | denom[N*8]  (~38.4 MB)
    float*    g     = (float*)d_ws;
    float*    e_s   = g + (size_t)n * HID;
    float*    e_d   = e_s + (size_t)n * NH;
    unsigned* m_enc = (unsigned*)(e_d + (size_t)n * NH);
    float*    denom = (float*)(m_enc + (size_t)n * NH);
    float*    out   = (float*)d_out;

    const int TB = 256;
    init_kernel<<<(n * HID + TB - 1) / TB, TB, 0, stream>>>(out, m_enc, denom, n);

    const int mtiles = (n + 15) / 16;
    gat_gemm_wmma<<<(mtiles + 1) / 2, TB, 0, stream>>>(vert, W, g, n);

    edot_kernel<<<(n * NH + TB - 1) / TB, TB, 0, stream>>>(g, a_src, a_dst, e_s, e_d, n);

    edge_max_kernel<<<(ne * NH + TB - 1) / TB, TB, 0, stream>>>(src, dst, e_s, e_d, m_enc, ne);
    edge_den_kernel<<<(ne * NH + TB - 1) / TB, TB, 0, stream>>>(src, dst, e_s, e_d, m_enc, denom, ne);
    edge_agg_kernel<<<(ne * 16 + TB - 1) / TB, TB, 0, stream>>>(src, dst, e_s, e_d, m_enc, denom, g, out, ne);

    elu_kernel<<<(n * HID + TB - 1) / TB, TB, 0, stream>>>(out, n * HID);
}